// GCNNodeAdaptiveExit_87608742904007
// MI455X (gfx1250) — compile-verified
//
#include <hip/hip_runtime.h>
#include <hip/hip_bf16.h>
#include <math.h>

#define NN   50000
#define EE   800000
#define HIDD 128
#define CHID 64
#define BSTRIDE 80   // LDS row stride (floats): keeps wave-halves on disjoint banks

typedef __attribute__((ext_vector_type(2))) float v2f;
typedef __attribute__((ext_vector_type(8))) float v8f;

__device__ __forceinline__ float gelu_exact(float v) {
    return 0.5f * v * (1.0f + erff(v * 0.70710678118654752f));
}

__device__ __forceinline__ unsigned hash_u32(unsigned x) {
    x ^= x >> 16; x *= 0x7feb352dU;
    x ^= x >> 15; x *= 0x846ca68bU;
    x ^= x >> 16;
    return x;
}

// Deterministic stand-in for jax.random gumbel noise: -log(-log(U)), U in [1e-10,1).
__device__ __forceinline__ float gumbel_noise(unsigned node, unsigned comp, unsigned layer) {
    unsigned h = hash_u32(node * 2u + comp + (layer + 1u) * 0x9E3779B9u);
    float u = fmaxf((float)(h >> 8) * (1.0f / 16777216.0f), 1e-10f);
    return -logf(-logf(u));
}

// ---------------------------------------------------------------------------
// f32 WMMA GEMM: C[M x Ncols] = A[M x K] @ B[K x Ncols] (+bias)(+act)
// Block = 8 waves = 128 rows; each wave owns 16 rows x 64 cols (4 accumulators,
// A fragment reused 4x). B panel (K x 64) staged in LDS, shared by all 8 waves.
// K <= 128, K % 4 == 0, Ncols % 64 == 0. Tail rows: clamped loads, guarded stores.
// act: 0 none, 1 exact GELU, 2 ReLU.
// ---------------------------------------------------------------------------
__global__ __launch_bounds__(256) void wmma_gemm_f32(
    const float* __restrict__ A, const float* __restrict__ Bm,
    const float* __restrict__ bias, float* __restrict__ C,
    int M, int K, int Ncols, int act)
{
    __shared__ float Bs[128 * BSTRIDE];

    const int tid  = threadIdx.x;
    const int wave = tid >> 5;
    const int lane = tid & 31;
    const int n0   = blockIdx.y * 64;
    const int m0   = blockIdx.x * 128 + wave * 16;

    // Cooperative stage of B[0:K, n0:n0+64] into LDS (float4 per thread-iter).
    for (int idx = tid; idx < K * 16; idx += 256) {
        const int row = idx >> 4;
        const int col = (idx & 15) << 2;
        *(float4*)&Bs[row * BSTRIDE + col] =
            *(const float4*)&Bm[(size_t)row * Ncols + n0 + col];
    }
    __syncthreads();

    const int r16  = lane & 15;
    const int koff = (lane >> 4) << 1;          // lanes 16-31 carry K+2,K+3
    long long arow = m0 + r16;
    if (arow >= M) arow = M - 1;                // clamp: keeps EXEC full for WMMA
    const float* ap = A + arow * (long long)K;

    v8f acc[4] = {v8f{}, v8f{}, v8f{}, v8f{}};
    for (int k0 = 0; k0 < K; k0 += 4) {
        const int ka = k0 + koff;
        v2f a;
        a.x = ap[ka];
        a.y = ap[ka + 1];
#pragma unroll
        for (int t = 0; t < 4; ++t) {
            v2f b;
            b.x = Bs[ka * BSTRIDE       + t * 16 + r16];
            b.y = Bs[(ka + 1) * BSTRIDE + t * 16 + r16];
            acc[t] = __builtin_amdgcn_wmma_f32_16x16x4_f32(
                false, a, false, b, (short)0, acc[t], false, false);
        }
    }

    // C/D layout: VGPR r -> rows (r | r+8) split across lane halves.
    const int srow0 = m0 + ((lane >> 4) << 3);
#pragma unroll
    for (int t = 0; t < 4; ++t) {
        const int col = n0 + t * 16 + r16;
        const float bv = bias ? bias[col] : 0.0f;
#pragma unroll
        for (int r = 0; r < 8; ++r) {
            const int row = srow0 + r;
            if (row < M) {
                float v = acc[t][r] + bv;
                if (act == 1)      v = gelu_exact(v);
                else if (act == 2) v = fmaxf(v, 0.0f);
                C[(long long)row * Ncols + col] = v;
            }
        }
    }
}

// ---------------------------------------------------------------------------
// Output init: z = 0, exit_layers = 3, active = 0
// ---------------------------------------------------------------------------
__global__ void init_out_kernel(float* fout, int* iout, int zcount, int n) {
    int idx = blockIdx.x * blockDim.x + threadIdx.x;
    if (idx >= zcount + n + 3) return;
    if (idx < zcount)          fout[idx] = 0.0f;
    else if (idx < zcount + n) iout[idx] = 3;
    else                       iout[idx] = 0;
}

__global__ void zero_int_kernel(int* p, int n) {
    int i = blockIdx.x * blockDim.x + threadIdx.x;
    if (i < n) p[i] = 0;
}

__global__ void deg_edges_kernel(const int* __restrict__ dst, int* __restrict__ deg, int e) {
    int i = blockIdx.x * blockDim.x + threadIdx.x;
    if (i < e) atomicAdd(&deg[dst[i]], 1);
}

__global__ void dinv_kernel(const int* __restrict__ deg, float* __restrict__ dinv, int n) {
    int i = blockIdx.x * blockDim.x + threadIdx.x;
    if (i < n) dinv[i] = rsqrtf((float)(deg[i] + 1));   // +1 self loop; always > 0
}

// Single-block exclusive prefix scan: off[0..n] from cnt[0..n-1].
__global__ __launch_bounds__(256) void scan_kernel(const int* __restrict__ cnt,
                                                   int* __restrict__ off, int n) {
    __shared__ int part[256];
    const int t = threadIdx.x;
    const int chunk = (n + 255) / 256;
    const int lo = t * chunk;
    const int hi = min(lo + chunk, n);
    int s = 0;
    for (int i = lo; i < hi; ++i) s += cnt[i];
    part[t] = s;
    __syncthreads();
    for (int d = 1; d < 256; d <<= 1) {
        int v = (t >= d) ? part[t - d] : 0;
        __syncthreads();
        part[t] += v;
        __syncthreads();
    }
    int run = (t == 0) ? 0 : part[t - 1];
    for (int i = lo; i < hi; ++i) { off[i] = run; run += cnt[i]; }
    if (t == 255) off[n] = run;
}

__global__ void copy_int_kernel(const int* __restrict__ a, int* __restrict__ b, int n) {
    int i = blockIdx.x * blockDim.x + threadIdx.x;
    if (i < n) b[i] = a[i];
}

// Bin edge ids by destination (order within bin resolved by sort below).
__global__ void fill_csr_kernel(const int* __restrict__ dst, int* __restrict__ cursor,
                                int* __restrict__ csr, int e) {
    int i = blockIdx.x * blockDim.x + threadIdx.x;
    if (i < e) {
        int p = atomicAdd(&cursor[dst[i]], 1);
        csr[p] = i;
    }
}

// Canonicalize each bin by ascending edge id -> bit-deterministic fp sums.
__global__ void sort_csr_kernel(const int* __restrict__ off, int* __restrict__ csr, int n) {
    int i = blockIdx.x * blockDim.x + threadIdx.x;
    if (i >= n) return;
    const int lo = off[i], hi = off[i + 1];
    for (int a = lo + 1; a < hi; ++a) {
        int v = csr[a];
        int b = a - 1;
        while (b >= lo && csr[b] > v) { csr[b + 1] = csr[b]; --b; }
        csr[b + 1] = v;
    }
}

// Replace edge ids with source node ids (one load per edge in the hot loop).
__global__ void csr_to_src_kernel(const int* __restrict__ srcp, int* __restrict__ csr, int e) {
    int i = blockIdx.x * blockDim.x + threadIdx.x;
    if (i < e) csr[i] = srcp[csr[i]];
}

// One wave per destination node. Each edge gather reads a contiguous 512B row
// of h[src] across the 32 lanes (coalesced, L2-resident). Fuses self-loop,
// bias and optional GELU; writes each output row exactly once.
__global__ __launch_bounds__(256) void aggregate_kernel(
    const float* __restrict__ h, const int* __restrict__ off,
    const int* __restrict__ csr_src, const float* __restrict__ dinv,
    const float* __restrict__ bias, float* __restrict__ out, int n, int act)
{
    const int node = blockIdx.x * 8 + (threadIdx.x >> 5);
    if (node >= n) return;
    const int lane = threadIdx.x & 31;
    const int c = lane * 4;
    const float di = dinv[node];

    float4 acc = make_float4(0.f, 0.f, 0.f, 0.f);
    const int lo = off[node], hi = off[node + 1];
    for (int e = lo; e < hi; ++e) {
        const int s = csr_src[e];
        const float w = dinv[s] * di;
        const float4 hs = *(const float4*)(h + (size_t)s * HIDD + c);
        acc.x += hs.x * w; acc.y += hs.y * w;
        acc.z += hs.z * w; acc.w += hs.w * w;
    }
    // self loop (norm = dinv[i]^2), then bias
    const float4 hv = *(const float4*)(h + (size_t)node * HIDD + c);
    const float d2 = di * di;
    acc.x += hv.x * d2 + bias[c + 0];
    acc.y += hv.y * d2 + bias[c + 1];
    acc.z += hv.z * d2 + bias[c + 2];
    acc.w += hv.w * d2 + bias[c + 3];
    if (act == 1) {
        acc.x = gelu_exact(acc.x); acc.y = gelu_exact(acc.y);
        acc.z = gelu_exact(acc.z); acc.w = gelu_exact(acc.w);
    }
    *(float4*)(out + (size_t)node * HIDD + c) = acc;
}

// Per-node exit decision. softmax((logits+g)/temp) argmax is invariant to the
// positive temperature, so temp_w never affects any output -> skipped.
__global__ void decide_kernel(const float* __restrict__ hid,
                              const float* __restrict__ w2, const float* __restrict__ b2,
                              int* __restrict__ exit_layers, int* __restrict__ active,
                              int layer, int n) {
    int i = blockIdx.x * blockDim.x + threadIdx.x;
    if (i >= n) return;
    const float* hp = hid + (long long)i * CHID;
    float l0 = b2[0], l1 = b2[1];
#pragma unroll 8
    for (int c = 0; c < CHID; ++c) {
        float hv = hp[c];
        l0 += hv * w2[c * 2 + 0];
        l1 += hv * w2[c * 2 + 1];
    }
    const float g0 = gumbel_noise((unsigned)i, 0u, (unsigned)layer);
    const float g1 = gumbel_noise((unsigned)i, 1u, (unsigned)layer);
    const bool dec  = (l1 + g1) > (l0 + g0);
    const bool cont = (exit_layers[i] == 3);
    if (cont) atomicAdd(&active[layer], 1);
    if (cont && dec) exit_layers[i] = layer;
}

// z += x for nodes that newly exited this layer (final layer also folds in the
// never-exited nodes, which receive the same layer-2 features).
__global__ void zaccum_kernel(const float* __restrict__ x, const int* __restrict__ exit_layers,
                              float* __restrict__ z, int layer, long long total) {
    long long idx = (long long)blockIdx.x * blockDim.x + threadIdx.x;
    if (idx >= total) return;
    const int el = exit_layers[(int)(idx >> 7)];
    const bool add = (layer == 2) ? (el >= 2) : (el == layer);
    if (add) z[idx] += x[idx];
}

extern "C" void kernel_launch(void* const* d_in, const int* in_sizes, int n_in,
                              void* d_out, int out_size, void* d_ws, size_t ws_size,
                              hipStream_t stream) {
    (void)in_sizes; (void)n_in; (void)out_size; (void)ws_size;

    const float* x         = (const float*)d_in[0];
    const int*   ei        = (const int*)d_in[1];
    const float* conv_w[3] = {(const float*)d_in[2], (const float*)d_in[4], (const float*)d_in[6]};
    const float* conv_b[3] = {(const float*)d_in[3], (const float*)d_in[5], (const float*)d_in[7]};
    const float* conf_w1   = (const float*)d_in[8];
    const float* conf_b1   = (const float*)d_in[9];
    const float* conf_w2   = (const float*)d_in[10];
    const float* conf_b2   = (const float*)d_in[11];
    // d_in[12] (temp_w) provably unused: positive temperature cancels in the argmax.

    const int N = NN, E = EE;
    const int* srcp = ei;
    const int* dstp = ei + E;

    float* zout  = (float*)d_out;
    int*   exitl = (int*)d_out + (size_t)N * HIDD;
    int*   activ = exitl + N;

    float* wsf    = (float*)d_ws;
    float* hbuf   = wsf;            wsf += (size_t)N * HIDD;
    float* xb0    = wsf;            wsf += (size_t)N * HIDD;
    float* xb1    = wsf;            wsf += (size_t)N * HIDD;
    float* hid    = wsf;            wsf += (size_t)N * CHID;
    float* dinv   = wsf;            wsf += N;
    int*   deg    = (int*)wsf;      wsf += N;
    int*   off    = (int*)wsf;      wsf += N + 1;
    int*   cursor = (int*)wsf;      wsf += N;
    int*   csr    = (int*)wsf;      wsf += E;

    const long long NH = (long long)N * HIDD;
    const dim3 b256(256);
    const dim3 gNH((unsigned)((NH + 255) / 256));
    const dim3 gN((unsigned)((N + 255) / 256));
    const dim3 gE((unsigned)((E + 255) / 256));

    // Re-initialize all outputs every launch (graph-replay safe, deterministic).
    {
        const long long tot = NH + N + 3;
        init_out_kernel<<<dim3((unsigned)((tot + 255) / 256)), b256, 0, stream>>>(
            zout, (int*)d_out, (int)NH, N);
    }

    // ---- Build sym-norm + CSR-by-destination (deterministic edge order) ----
    zero_int_kernel<<<gN, b256, 0, stream>>>(deg, N);
    deg_edges_kernel<<<gE, b256, 0, stream>>>(dstp, deg, E);
    dinv_kernel<<<gN, b256, 0, stream>>>(deg, dinv, N);
    scan_kernel<<<dim3(1), b256, 0, stream>>>(deg, off, N);
    copy_int_kernel<<<gN, b256, 0, stream>>>(off, cursor, N);
    fill_csr_kernel<<<gE, b256, 0, stream>>>(dstp, cursor, csr, E);
    sort_csr_kernel<<<gN, b256, 0, stream>>>(off, csr, N);
    csr_to_src_kernel<<<gE, b256, 0, stream>>>(srcp, csr, E);

    const float* xin = x;
    float* xbufs[2] = {xb0, xb1};
    const dim3 gemmBlk(256);
    const unsigned mblocks = (unsigned)((N + 127) / 128);   // 391

    for (int layer = 0; layer < 3; ++layer) {
        float* xout = xbufs[layer & 1];

        // h = xin @ conv_w[layer]   (f32 WMMA, LDS-staged B, 16x64 per wave)
        wmma_gemm_f32<<<dim3(mblocks, HIDD / 64), gemmBlk, 0, stream>>>(
            xin, conv_w[layer], nullptr, hbuf, N, HIDD, HIDD, /*act=*/0);

        // xout = sym-norm aggregate (gather) + self loop + bias (+ GELU)
        aggregate_kernel<<<dim3((unsigned)((N + 7) / 8)), b256, 0, stream>>>(
            hbuf, off, csr, dinv, conv_b[layer], xout, N, (layer < 2) ? 1 : 0);

        // Confidence hidden layer: relu(xout @ conf_w1 + conf_b1)  (WMMA)
        wmma_gemm_f32<<<dim3(mblocks, CHID / 64), gemmBlk, 0, stream>>>(
            xout, conf_w1, conf_b1, hid, N, HIDD, CHID, /*act=*/2);

        decide_kernel<<<gN, b256, 0, stream>>>(hid, conf_w2, conf_b2, exitl, activ, layer, N);
        zaccum_kernel<<<gNH, b256, 0, stream>>>(xout, exitl, zout, layer, NH);

        xin = xout;
    }
}